// TF_linear_att_49383533969911
// MI455X (gfx1250) — compile-verified
//
#include <hip/hip_runtime.h>
#include <math.h>

// fp32 WMMA operand types: A/B 16x4 & 4x16 f32 = 64 elems / 32 lanes = 2 VGPRs,
// C/D 16x16 f32 = 256 elems / 32 lanes = 8 VGPRs.
typedef __attribute__((ext_vector_type(2))) float v2f;
typedef __attribute__((ext_vector_type(8))) float v8f;

#define BATCH 16
#define NCTX  4096
#define DF    32
#define ROWSTRIDE (2 * DF)                       // 64 floats per x row
#define SPLIT 32                                 // K-dim split across blocks
#define WAVES 8                                  // 256-thread block = 8 wave32
#define ROWS_PER_BLOCK (NCTX / SPLIT)            // 128
#define ROWS_PER_WAVE  (ROWS_PER_BLOCK / WAVES)  // 16 (4 WMMA K-steps of 4)

// Kernel 1: per (batch, split) block compute partial M = X2^T @ X1 over a
// 128-row slice with V_WMMA_F32_16X16X4_F32, reduce the 8 waves in LDS
// (ds_add_f32), and write the 32x32 partial to workspace.
__global__ __launch_bounds__(256) void TF_linatt_mm_partial(
    const float* __restrict__ x, float* __restrict__ partial) {
  __shared__ float M_s[DF][DF + 1];  // +1 pad to spread LDS banks

  const int b    = blockIdx.x;
  const int sp   = blockIdx.y;
  const int tid  = threadIdx.x;
  const int lane = tid & 31;
  const int wid  = tid >> 5;

  for (int i = tid; i < DF * (DF + 1); i += 256) ((float*)M_s)[i] = 0.0f;
  __syncthreads();

  const float* xb = x + (size_t)b * (NCTX + 1) * ROWSTRIDE;
  const int i0 = sp * ROWS_PER_BLOCK + wid * ROWS_PER_WAVE;

  // WMMA f32 16x16x4 operand lane mapping (ISA 7.12.2):
  //   A: lane m = lane&15, VGPR v holds K = 2*(lane>>4) + v
  //   B: lane n = lane&15, VGPR v holds K = 2*(lane>>4) + v  (symmetric)
  const int m  = lane & 15;
  const int kk = (lane >> 4) << 1;  // 0 or 2

  v8f acc00 = {}; v8f acc01 = {}; v8f acc10 = {}; v8f acc11 = {};

  for (int i = i0; i < i0 + ROWS_PER_WAVE; i += 4) {
    const float* r0 = xb + (size_t)(i + kk) * ROWSTRIDE;  // row i+kk
    const float* r1 = r0 + ROWSTRIDE;                     // row i+kk+1
    v2f a0, a1, b0v, b1v;
    // A = X2^T : A[c, k] = x[b, i+k, DF + c]
    a0.x = r0[DF + m];      a0.y = r1[DF + m];
    a1.x = r0[DF + 16 + m]; a1.y = r1[DF + 16 + m];
    // B = X1   : B[k, j] = x[b, i+k, j]
    b0v.x = r0[m];          b0v.y = r1[m];
    b1v.x = r0[16 + m];     b1v.y = r1[16 + m];

    acc00 = __builtin_amdgcn_wmma_f32_16x16x4_f32(false, a0, false, b0v,
                                                  (short)0, acc00, false, false);
    acc01 = __builtin_amdgcn_wmma_f32_16x16x4_f32(false, a0, false, b1v,
                                                  (short)0, acc01, false, false);
    acc10 = __builtin_amdgcn_wmma_f32_16x16x4_f32(false, a1, false, b0v,
                                                  (short)0, acc10, false, false);
    acc11 = __builtin_amdgcn_wmma_f32_16x16x4_f32(false, a1, false, b1v,
                                                  (short)0, acc11, false, false);
  }

  // C/D layout: VGPR e, lanes 0-15 -> M=e, lanes 16-31 -> M=8+e; N=lane&15.
  const int rbase = (lane >> 4) << 3;  // 0 or 8
  const int col   = lane & 15;
#pragma unroll
  for (int e = 0; e < 8; ++e) {
    atomicAdd(&M_s[rbase + e][col],           acc00[e]);
    atomicAdd(&M_s[rbase + e][16 + col],      acc01[e]);
    atomicAdd(&M_s[16 + rbase + e][col],      acc10[e]);
    atomicAdd(&M_s[16 + rbase + e][16 + col], acc11[e]);
  }
  __syncthreads();

  float* outp = partial + ((size_t)b * SPLIT + sp) * (DF * DF);
  for (int idx = tid; idx < DF * DF; idx += 256)
    outp[idx] = M_s[idx >> 5][idx & 31];
}

// Kernel 2: per batch, sum the SPLIT partial M matrices, then the cheap chain
//   q = W1 @ x[b,n,:d];  t = M @ q;  out = x[b,n,d:] + (W0 @ t)/n
__global__ __launch_bounds__(256) void TF_linatt_finalize(
    const float* __restrict__ x, const float* __restrict__ W0,
    const float* __restrict__ W1, const float* __restrict__ partial,
    float* __restrict__ out) {
  __shared__ float M_s[DF][DF];
  __shared__ float q_s[DF];
  __shared__ float t_s[DF];

  const int b   = blockIdx.x;
  const int tid = threadIdx.x;

  for (int idx = tid; idx < DF * DF; idx += 256) {
    float s = 0.0f;
    const float* p = partial + (size_t)b * SPLIT * (DF * DF) + idx;
#pragma unroll 4
    for (int sp = 0; sp < SPLIT; ++sp) s += p[(size_t)sp * DF * DF];
    M_s[idx >> 5][idx & 31] = s;
  }
  const float* xn = x + ((size_t)b * (NCTX + 1) + NCTX) * ROWSTRIDE;
  __syncthreads();

  if (tid < DF) {
    float q = 0.0f;
#pragma unroll
    for (int mm = 0; mm < DF; ++mm) q += W1[tid * DF + mm] * xn[mm];
    q_s[tid] = q;
  }
  __syncthreads();
  if (tid < DF) {
    float t = 0.0f;
#pragma unroll
    for (int k = 0; k < DF; ++k) t += M_s[tid][k] * q_s[k];
    t_s[tid] = t;
  }
  __syncthreads();
  if (tid < DF) {
    float u = 0.0f;
#pragma unroll
    for (int c = 0; c < DF; ++c) u += W0[tid * DF + c] * t_s[c];
    out[b * DF + tid] = xn[DF + tid] + u * (1.0f / (float)NCTX);
  }
}

extern "C" void kernel_launch(void* const* d_in, const int* in_sizes, int n_in,
                              void* d_out, int out_size, void* d_ws, size_t ws_size,
                              hipStream_t stream) {
  const float* x  = (const float*)d_in[0];  // (B, n+1, 2d) fp32
  const float* W0 = (const float*)d_in[1];  // (d, d)
  const float* W1 = (const float*)d_in[2];  // (d, d)
  float* out      = (float*)d_out;          // (B, d)
  float* partial  = (float*)d_ws;           // B*SPLIT*32*32 floats = 2 MiB

  dim3 grid(BATCH, SPLIT);
  TF_linatt_mm_partial<<<grid, 256, 0, stream>>>(x, partial);
  TF_linatt_finalize<<<BATCH, 256, 0, stream>>>(x, W0, W1, partial, out);
}